// RecurrentNeuralNetwork_84567906059001
// MI455X (gfx1250) — compile-verified
//
#include <hip/hip_runtime.h>
#include <hip/hip_bf16.h>
#include <math.h>

// ---------------- problem constants ----------------
#define N_IN   128
#define N_HID  1024
#define N_OUT  64
#define BB     128
#define TT     256
#define SIGMA_NEU 0.05f

#define K8_HH  (N_HID / 8)   // 128 K-blocks of 8 for hidden GEMM
#define K8_IN  (N_IN  / 8)   // 16
// padded stride for row-major h in LDS (1028 mod 64 == 4 -> conflict-free col gathers)
#define HSTRIDE 1028

#define OUT_OUT_OFF (BB * TT * N_HID)                    // 33554432
#define OUT_FIN_OFF (OUT_OUT_OFF + BB * TT * N_OUT)      // 35651584

typedef float v2f __attribute__((ext_vector_type(2)));
typedef float v8f __attribute__((ext_vector_type(8)));
typedef float f4  __attribute__((ext_vector_type(4)));

static __device__ __forceinline__ v8f wmma_f32(v2f a, v2f b, v8f c) {
  // V_WMMA_F32_16X16X4_F32 : D = A(16x4,f32) * B(4x16,f32) + C(16x16,f32)
  return __builtin_amdgcn_wmma_f32_16x16x4_f32(false, a, false, b, (short)0, c,
                                               false, false);
}
static __device__ __forceinline__ v2f lo2(f4 v) { v2f r; r.x = v.x; r.y = v.y; return r; }
static __device__ __forceinline__ v2f hi2(f4 v) { v2f r; r.x = v.z; r.y = v.w; return r; }

// gfx1250 hardware tanh (V_TANH_F32) when available; ocml fallback otherwise.
static __device__ __forceinline__ float fast_tanh(float x) {
#if __has_builtin(__builtin_amdgcn_tanhf)
  return __builtin_amdgcn_tanhf(x);
#else
  return tanhf(x);
#endif
}

// ---------------------------------------------------------------------------
// Pack weight matrix w[n_units][K] (row-major) into WMMA B-operand order:
// dst[(((ct*K8 + k8)*32 + l)*4 + q)] = w[col][k]
//   col = ct*16 + (l&15), half = l>>4, kk = ((q>>1)<<2)|(half<<1)|(q&1),
//   k = k8*8 + kk.
// A lane-l b128 load at block (ct,k8) then yields exactly the two v2f operands
// for the WMMAs at K = 8*k8 and K = 8*k8+4.
// ---------------------------------------------------------------------------
__global__ void pack_b_operand(const float* __restrict__ src,
                               float* __restrict__ dst, int K, int total) {
  int idx = blockIdx.x * blockDim.x + threadIdx.x;
  if (idx >= total) return;
  int q    = idx & 3;
  int l    = (idx >> 2) & 31;
  int rest = idx >> 7;
  int nk8  = K >> 3;
  int k8   = rest % nk8;
  int ct   = rest / nk8;
  int col  = ct * 16 + (l & 15);
  int half = l >> 4;
  int kk   = ((q >> 1) << 2) | (half << 1) | (q & 1);
  int k    = k8 * 8 + kk;
  dst[idx] = src[col * K + k];
}

// ---------------------------------------------------------------------------
// Persistent RNN scan: one workgroup per batch tile of 16, 32 wave32s.
// ---------------------------------------------------------------------------
__global__ __launch_bounds__(1024, 1)
void rnn_scan_kernel(const float* __restrict__ xin,     // [B,T,N_IN]
                     const float* __restrict__ h0,      // [B,N_HID]
                     const float* __restrict__ bias,    // [N_HID]
                     const float* __restrict__ alpha,   // [N_HID]
                     const float* __restrict__ noise,   // [B,N_HID]
                     const int*   __restrict__ pt_ptr,  // scalar
                     const float* __restrict__ WinP,    // packed B operands
                     const float* __restrict__ WhhP,    // packed B operands
                     const float* __restrict__ WoutP,   // packed B operands
                     float* __restrict__ out) {
  extern __shared__ float smem[];
  float* h_lds  = smem;                    // 16 x HSTRIDE   (row-major h)
  float* a_pack = smem + 16 * HSTRIDE;     // 16384 floats   (tanh(h), A-op packed)
  float* x_pack = a_pack + 16 * N_HID;     // 2048 floats    (x_t, A-op packed)

  const int tid   = threadIdx.x;
  const int lane  = tid & 31;
  const int wave  = tid >> 5;             // 32 waves
  const int bbase = blockIdx.x * 16;
  const int pt    = pt_ptr[0];

  const int mrow  = lane & 15;
  const int koff  = (lane >> 4) << 1;     // lanes 16-31 carry K+2,K+3
  const int Mbase = (lane < 16) ? 0 : 8;

  // each wave owns two 16-wide hidden column tiles
  const int tile0 = wave * 2;
  const int col0  = tile0 * 16 + mrow;
  const int col1  = col0 + 16;
  const float al0 = alpha[col0], al1 = alpha[col1];
  const float bi0 = bias[col0],  bi1 = bias[col1];
  const float ns0 = SIGMA_NEU * sqrtf(al0);
  const float ns1 = SIGMA_NEU * sqrtf(al1);

  // loop-invariant operand base pointers (imm offsets cover whole K range)
  const float* ap   = a_pack + lane * 4;
  const float* xp   = x_pack + lane * 4;
  const float* bh0  = WhhP + (size_t)tile0 * (K8_HH * 128) + lane * 4;
  const float* bh1  = bh0 + (K8_HH * 128);
  const float* bin0 = WinP + (size_t)tile0 * (K8_IN * 128) + lane * 4;
  const float* bin1 = bin0 + (K8_IN * 128);
  const float* bo   = WoutP + (size_t)wave * (K8_HH * 128) + lane * 4; // wave<4

  // load initial hidden state into LDS (float4-coalesced)
  for (int f = tid; f < 4096; f += 1024) {
    int m = f >> 8, j = (f & 255) << 2;
    *(f4*)&h_lds[m * HSTRIDE + j] = *(const f4*)&h0[(bbase + m) * N_HID + j];
  }
  __syncthreads();

  for (int t = 0; t < TT; ++t) {
    // ---- Phase A: a_pack = packed tanh(h); x_pack = packed x_t ----------
    // mapping m = f&15 keeps LDS reads/writes bank-spread
    for (int f = tid; f < 4096; f += 1024) {
      int m = f & 15, j = (f >> 4) << 2;       // j multiple of 4
      f4 h = *(f4*)&h_lds[m * HSTRIDE + j];
      f4 a;
      a.x = fast_tanh(h.x); a.y = fast_tanh(h.y);
      a.z = fast_tanh(h.z); a.w = fast_tanh(h.w);
      int k8 = j >> 3, qo = (j & 4) >> 1;
      v2f s0; s0.x = a.x; s0.y = a.y;
      v2f s1; s1.x = a.z; s1.y = a.w;
      *(v2f*)&a_pack[(k8 * 32 + m) * 4 + qo]      = s0;  // half 0
      *(v2f*)&a_pack[(k8 * 32 + 16 + m) * 4 + qo] = s1;  // half 1
    }
    if (tid < 512) {  // 16 rows x 32 float4, coalesced global reads
      const float* xrow =
          &xin[(size_t)(bbase + (tid >> 5)) * TT * N_IN + (size_t)t * N_IN];
      int j = (tid & 31) << 2;
      f4 x = *(const f4*)&xrow[j];
      if (t + 1 < TT) __builtin_prefetch(xrow + N_IN + j, 0, 1);
      int m = tid >> 5;
      int k8 = j >> 3, qo = (j & 4) >> 1;
      v2f s0; s0.x = x.x; s0.y = x.y;
      v2f s1; s1.x = x.z; s1.y = x.w;
      *(v2f*)&x_pack[(k8 * 32 + m) * 4 + qo]      = s0;
      *(v2f*)&x_pack[(k8 * 32 + 16 + m) * 4 + qo] = s1;
    }
    __syncthreads();

    // ---- Phase B: tmp = x_t @ Win^T + act @ Whh^T (two tiles per wave) --
    v8f acc0 = {}, acc1 = {};

#pragma unroll 4
    for (int k8 = 0; k8 < K8_IN; ++k8) {
      f4 a  = *(const f4*)(xp   + (k8 << 7));
      f4 b0 = *(const f4*)(bin0 + (k8 << 7));
      f4 b1 = *(const f4*)(bin1 + (k8 << 7));
      acc0 = wmma_f32(lo2(a), lo2(b0), acc0);
      acc1 = wmma_f32(lo2(a), lo2(b1), acc1);
      acc0 = wmma_f32(hi2(a), hi2(b0), acc0);
      acc1 = wmma_f32(hi2(a), hi2(b1), acc1);
    }

#pragma unroll 4
    for (int k8 = 0; k8 < K8_HH; ++k8) {
      f4 a  = *(const f4*)(ap  + (k8 << 7));
      f4 b0 = *(const f4*)(bh0 + (k8 << 7));
      f4 b1 = *(const f4*)(bh1 + (k8 << 7));
      acc0 = wmma_f32(lo2(a), lo2(b0), acc0);
      acc1 = wmma_f32(lo2(a), lo2(b1), acc1);
      acc0 = wmma_f32(hi2(a), hi2(b0), acc0);
      acc1 = wmma_f32(hi2(a), hi2(b1), acc1);
    }

    // ---- Phase C: h_new = (1-a)h + a(tmp+b) (+ noise at t==pt) ----------
#pragma unroll
    for (int r = 0; r < 8; ++r) {
      int M = Mbase + r;
      float h0v = h_lds[M * HSTRIDE + col0];
      float h1v = h_lds[M * HSTRIDE + col1];
      float hn0 = (1.0f - al0) * h0v + al0 * (acc0[r] + bi0);
      float hn1 = (1.0f - al1) * h1v + al1 * (acc1[r] + bi1);
      if (t == pt) {
        hn0 += noise[(bbase + M) * N_HID + col0] * ns0;
        hn1 += noise[(bbase + M) * N_HID + col1] * ns1;
      }
      h_lds[M * HSTRIDE + col0] = hn0;
      h_lds[M * HSTRIDE + col1] = hn1;
    }
    __syncthreads();

    // ---- Phase D: out-projection (waves 0-3) || hidden_list writeback ---
    if (wave < 4) {
      v8f ao = {};
      const int ocol = wave * 16 + mrow;
      const float* hrow = h_lds + mrow * HSTRIDE + koff;
#pragma unroll 4
      for (int k8 = 0; k8 < K8_HH; ++k8) {
        v2f a0; a0.x = hrow[k8 * 8];     a0.y = hrow[k8 * 8 + 1];
        v2f a1; a1.x = hrow[k8 * 8 + 4]; a1.y = hrow[k8 * 8 + 5];
        f4  b  = *(const f4*)(bo + (k8 << 7));
        ao = wmma_f32(a0, lo2(b), ao);
        ao = wmma_f32(a1, hi2(b), ao);
      }
#pragma unroll
      for (int r = 0; r < 8; ++r) {
        int M = Mbase + r;
        out[(size_t)OUT_OUT_OFF + (size_t)(bbase + M) * TT * N_OUT +
            (size_t)t * N_OUT + ocol] = ao[r];
      }
    } else {
      for (int f = tid - 128; f < 4096; f += 896) {
        int m = f >> 8, j = (f & 255) << 2;
        *(f4*)&out[(size_t)(bbase + m) * TT * N_HID + (size_t)t * N_HID + j] =
            *(f4*)&h_lds[m * HSTRIDE + j];
      }
    }
    __syncthreads();
  }

  // ---- h_final ----------------------------------------------------------
  for (int f = tid; f < 4096; f += 1024) {
    int m = f >> 8, j = (f & 255) << 2;
    *(f4*)&out[(size_t)OUT_FIN_OFF + (size_t)(bbase + m) * N_HID + j] =
        *(f4*)&h_lds[m * HSTRIDE + j];
  }
}

extern "C" void kernel_launch(void* const* d_in, const int* in_sizes, int n_in,
                              void* d_out, int out_size, void* d_ws, size_t ws_size,
                              hipStream_t stream) {
  const float* xin   = (const float*)d_in[0];
  const float* h0    = (const float*)d_in[1];
  const float* w_in  = (const float*)d_in[2];
  const float* w_hh  = (const float*)d_in[3];
  const float* bias  = (const float*)d_in[4];
  const float* w_out = (const float*)d_in[5];
  const float* alpha = (const float*)d_in[6];
  const float* noise = (const float*)d_in[7];
  const int*   ptim  = (const int*)d_in[8];
  float* out = (float*)d_out;

  // workspace: packed B-operand weights
  float* WinP  = (float*)d_ws;             // 128*1024
  float* WhhP  = WinP + N_IN * N_HID;      // 1024*1024
  float* WoutP = WhhP + N_HID * N_HID;     // 1024*64

  pack_b_operand<<<(N_IN * N_HID + 255) / 256, 256, 0, stream>>>(
      w_in, WinP, N_IN, N_IN * N_HID);
  pack_b_operand<<<(N_HID * N_HID + 255) / 256, 256, 0, stream>>>(
      w_hh, WhhP, N_HID, N_HID * N_HID);
  pack_b_operand<<<(N_HID * N_OUT + 255) / 256, 256, 0, stream>>>(
      w_out, WoutP, N_HID, N_HID * N_OUT);

  const size_t lds_bytes =
      (size_t)(16 * HSTRIDE + 16 * N_HID + 2048) * sizeof(float);  // ~136 KB
  (void)hipFuncSetAttribute((const void*)rnn_scan_kernel,
                            hipFuncAttributeMaxDynamicSharedMemorySize,
                            (int)lds_bytes);
  rnn_scan_kernel<<<BB / 16, 1024, lds_bytes, stream>>>(
      xin, h0, bias, alpha, noise, ptim, WinP, WhhP, WoutP, out);
}